// TrivialUpdater_68839735820395
// MI455X (gfx1250) — compile-verified
//
#include <hip/hip_runtime.h>
#include <stdint.h>

// Problem constants from the reference: B=2, S=2048, V=32000, D=4096, k=8.
#define TOPK    8
#define VOCAB   32000
#define DIM     4096
#define BLOCK   256
#define NPOS    (2 * 2048)
#define CHUNK   1024                 // D-elements per TDM gather chunk
#define NCHUNK  (DIM / CHUNK)        // 4

typedef __attribute__((ext_vector_type(4))) unsigned int u32x4;
typedef __attribute__((ext_vector_type(8))) int          i32x8;
typedef __attribute__((ext_vector_type(4))) int          i32x4;
typedef __attribute__((ext_vector_type(4))) float        f32x4;

// Insert (v,idx) into a descending-sorted top-8 kept in registers.
__device__ __forceinline__ void topk_insert(float v, int idx,
                                            float (&tv)[TOPK], int (&ti)[TOPK]) {
  if (v <= tv[TOPK - 1]) return;
  tv[TOPK - 1] = v;
  ti[TOPK - 1] = idx;
#pragma unroll
  for (int j = TOPK - 1; j > 0; --j) {
    if (tv[j] > tv[j - 1]) {
      float fv = tv[j]; tv[j] = tv[j - 1]; tv[j - 1] = fv;
      int   fi = ti[j]; ti[j] = ti[j - 1]; ti[j - 1] = fi;
    }
  }
}

// Issue one TDM gather: 8 rows (16-bit indices) x CHUNK f32 elements -> LDS.
// D# layout per CDNA5 ISA (gather mode).
__device__ __forceinline__ void tdm_gather_issue(uint64_t gaddr, uint32_t ldsaddr,
                                                 int p01, int p23, int p45, int p67) {
  u32x4 g0;
  g0[0] = 0x80000001u;                                       // count=1 | 16b idx | gather
  g0[1] = ldsaddr;                                           // LDS dest (bytes)
  g0[2] = (uint32_t)gaddr;                                   // global_addr[31:0]
  g0[3] = ((uint32_t)((gaddr >> 32) & 0x01FFFFFFull)) | (2u << 30);  // addr hi | type=2

  i32x8 g1;
  g1[0] = (int)(2u << 16);                                   // wg_mask=0, data_size=4B
  g1[1] = (int)(((uint32_t)DIM & 0xFFFFu) << 16);            // tensor_dim0 = DIM (lo16)
  g1[2] = (int)(((uint32_t)VOCAB & 0xFFFFu) << 16);          // dim0 hi=0 | tensor_dim1 lo
  g1[3] = (int)(((uint32_t)CHUNK & 0xFFFFu) << 16);          // dim1 hi=0 | tile_dim0=CHUNK
  g1[4] = TOPK;                                              // tile_dim1 = #valid indices
  g1[5] = DIM;                                               // tensor_dim0_stride lo32
  g1[6] = 0;
  g1[7] = 0;

  i32x4 g2; g2[0] = p01; g2[1] = p23; g2[2] = p45; g2[3] = p67;  // row indices 0..7
  i32x4 g3 = {0, 0, 0, 0};
  i32x8 gx = {0, 0, 0, 0, 0, 0, 0, 0};

  __builtin_amdgcn_tensor_load_to_lds(g0, g1, g2, g3, gx, 0);
}

__global__ __launch_bounds__(BLOCK) void topk_softmax_gather_kernel(
    const float* __restrict__ logits,   // (NPOS, VOCAB)
    const float* __restrict__ emb,      // (VOCAB, DIM)
    float* __restrict__ out)            // (NPOS, DIM)
{
  // 64 KB staging region: double-buffered TDM destination (2 x 8 x CHUNK f32).
  // The top-k merge scratch (16 KB) aliases into it -- it is dead before the
  // first TDM write can land.
  __shared__ __align__(16) float s_stage[2 * TOPK * CHUNK];
  __shared__ float s_prob[TOPK];
  __shared__ int   s_widx[TOPK];

  float* s_val = s_stage;                                 // BLOCK*TOPK floats (8 KB)
  int*   s_idx = (int*)(s_stage + BLOCK * TOPK);          // BLOCK*TOPK ints   (8 KB)

  const int pos = blockIdx.x;
  const int tid = threadIdx.x;

  // ---- Phase 1: streaming scan of logits (non-temporal: zero reuse, keep L2
  // capacity for embedding rows), per-thread top-8 ----
  float tv[TOPK];
  int   ti[TOPK];
#pragma unroll
  for (int j = 0; j < TOPK; ++j) { tv[j] = -3.4e38f; ti[j] = 0; }

  const f32x4* l4 = reinterpret_cast<const f32x4*>(logits + (size_t)pos * VOCAB);
  for (int i = tid; i < VOCAB / 4; i += BLOCK) {
    f32x4 c = __builtin_nontemporal_load(l4 + i);
    topk_insert(c.x, 4 * i + 0, tv, ti);
    topk_insert(c.y, 4 * i + 1, tv, ti);
    topk_insert(c.z, 4 * i + 2, tv, ti);
    topk_insert(c.w, 4 * i + 3, tv, ti);
  }
#pragma unroll
  for (int j = 0; j < TOPK; ++j) {
    s_val[tid * TOPK + j] = tv[j];
    s_idx[tid * TOPK + j] = ti[j];
  }

  // ---- Phase 2: LDS tree merge of sorted 8-lists ----
  for (int stride = BLOCK / 2; stride >= 1; stride >>= 1) {
    __syncthreads();
    if (tid < stride) {
      const int a = tid * TOPK;
      const int b = (tid + stride) * TOPK;
      int ia = 0, ib = 0;
      float ov[TOPK]; int oi[TOPK];
#pragma unroll
      for (int j = 0; j < TOPK; ++j) {
        float va = s_val[a + ia];
        float vb = s_val[b + ib];
        bool  ta = (va >= vb);
        ov[j] = ta ? va : vb;
        oi[j] = ta ? s_idx[a + ia] : s_idx[b + ib];
        ia += ta ? 1 : 0;
        ib += ta ? 0 : 1;
      }
#pragma unroll
      for (int j = 0; j < TOPK; ++j) { s_val[a + j] = ov[j]; s_idx[a + j] = oi[j]; }
    }
  }
  __syncthreads();

  // ---- Phase 3: softmax + move winners out of the aliased scratch ----
  if (tid == 0) {
    float m = s_val[0];                      // sorted desc -> max is [0]
    float e[TOPK];
    float sum = 0.f;
#pragma unroll
    for (int j = 0; j < TOPK; ++j) { e[j] = expf(s_val[j] - m); sum += e[j]; }
    float inv = 1.f / sum;
#pragma unroll
    for (int j = 0; j < TOPK; ++j) { s_prob[j] = e[j] * inv; s_widx[j] = s_idx[j]; }
  }
  __syncthreads();   // s_stage now dead as merge scratch; safe for TDM writes

  // Uniform (scalar) row indices for the descriptor, packed as 16-bit pairs.
  const int p01 = (__builtin_amdgcn_readfirstlane(s_widx[0]) & 0xFFFF) |
                  (__builtin_amdgcn_readfirstlane(s_widx[1]) << 16);
  const int p23 = (__builtin_amdgcn_readfirstlane(s_widx[2]) & 0xFFFF) |
                  (__builtin_amdgcn_readfirstlane(s_widx[3]) << 16);
  const int p45 = (__builtin_amdgcn_readfirstlane(s_widx[4]) & 0xFFFF) |
                  (__builtin_amdgcn_readfirstlane(s_widx[5]) << 16);
  const int p67 = (__builtin_amdgcn_readfirstlane(s_widx[6]) & 0xFFFF) |
                  (__builtin_amdgcn_readfirstlane(s_widx[7]) << 16);

  const uint64_t gbase     = (uint64_t)(uintptr_t)emb;
  const uint32_t lds_base  = (uint32_t)(uintptr_t)(&s_stage[0]);
  const uint32_t buf_bytes = TOPK * CHUNK * 4;   // 32 KB per buffer

  float p[TOPK];
#pragma unroll
  for (int j = 0; j < TOPK; ++j) p[j] = s_prob[j];

  // ---- Phase 4+5: double-buffered TDM gather pipeline ----
  // Wave 0 owns the TDM queue (its TENSORcnt tracks ops in order).
  if (tid < 32) {
    tdm_gather_issue(gbase,                 lds_base,             p01, p23, p45, p67);
    tdm_gather_issue(gbase + CHUNK * 4ull,  lds_base + buf_bytes, p01, p23, p45, p67);
  }

  float* op = out + (size_t)pos * DIM;
#pragma unroll
  for (int c = 0; c < NCHUNK; ++c) {
    if (tid < 32) {
      if (c < NCHUNK - 1) __builtin_amdgcn_s_wait_tensorcnt(1);  // chunk c landed
      else                __builtin_amdgcn_s_wait_tensorcnt(0);  // last chunk
    }
    __syncthreads();                       // chunk c visible to all waves

    const float* buf = s_stage + (c & 1) * (TOPK * CHUNK);
    const int d = tid * 4;                 // 256 threads x float4 == CHUNK
    f32x4 acc = {0.f, 0.f, 0.f, 0.f};
#pragma unroll
    for (int k = 0; k < TOPK; ++k) {
      f32x4 e = *reinterpret_cast<const f32x4*>(&buf[k * CHUNK + d]);
      acc.x += p[k] * e.x;
      acc.y += p[k] * e.y;
      acc.z += p[k] * e.z;
      acc.w += p[k] * e.w;
    }
    // Output has zero reuse: non-temporal store, don't displace embedding L2 lines.
    __builtin_nontemporal_store(acc, reinterpret_cast<f32x4*>(&op[c * CHUNK + d]));

    __syncthreads();                       // buffer (c&1) free for reuse
    if (c + 2 < NCHUNK && tid < 32) {
      tdm_gather_issue(gbase + (uint64_t)(c + 2) * (CHUNK * 4ull),
                       lds_base + (uint32_t)(c & 1) * buf_bytes,
                       p01, p23, p45, p67);
    }
  }
}

extern "C" void kernel_launch(void* const* d_in, const int* in_sizes, int n_in,
                              void* d_out, int out_size, void* d_ws, size_t ws_size,
                              hipStream_t stream) {
  const float* logits = (const float*)d_in[0];        // (2,2048,32000) f32
  // d_in[1] = prev_inputs: unused by the reference computation.
  const float* emb    = (const float*)d_in[2];        // (32000,4096) f32
  float* out          = (float*)d_out;                // (2,2048,4096) f32
  (void)in_sizes; (void)n_in; (void)out_size; (void)d_ws; (void)ws_size;

  topk_softmax_gather_kernel<<<dim3(NPOS), dim3(BLOCK), 0, stream>>>(logits, emb, out);
}